// SimplifiedRepetitiveInterestTracker_20031727469030
// MI455X (gfx1250) — compile-verified
//
#include <hip/hip_runtime.h>
#include <math.h>

typedef __attribute__((ext_vector_type(2))) float v2f;
typedef __attribute__((ext_vector_type(8))) float v8f;

#define BB 96
#define NN 512
#define MM 512
#define DD 32

// ---------------------------------------------------------------------------
// Kernel 1: cost_cm[b][j][i] = || seq1[b,i,:] - seq2[b,j,:] ||_2  (clamped)
// Stored COLUMN-major over (i,j): index = b*N*M + j*N + i, so the DP kernel
// (thread = column j) streams contiguous addresses down its column.
// One wave32 per 16x16 (i,j) tile; K=32 via 8x v_wmma_f32_16x16x4_f32.
// ---------------------------------------------------------------------------
__global__ __launch_bounds__(256) void dtw_cost_kernel(
    const float* __restrict__ seq1,
    const float* __restrict__ seq2,
    float* __restrict__ cost)
{
    const int WAVES = 8;
    const int wave = threadIdx.x >> 5;
    const int lane = threadIdx.x & 31;

    // tile id: 32x32 tiles per batch, 96 batches -> 98304 waves
    long gwave = (long)blockIdx.x * WAVES + wave;
    const int tj = (int)(gwave & 31);
    const int ti = (int)((gwave >> 5) & 31);
    const int b  = (int)(gwave >> 10);

    const float* s1 = seq1 + ((size_t)b * NN + (size_t)ti * 16) * DD; // 16 rows of A
    const float* s2 = seq2 + ((size_t)b * MM + (size_t)tj * 16) * DD; // 16 rows of B (cols of B^T)

    // --- row norms into LDS: lanes 0-15 -> sq1 rows, lanes 16-31 -> sq2 rows
    __shared__ float sq[WAVES][2][16];
    {
        const int r = lane & 15;
        const float* p = (lane < 16) ? (s1 + r * DD) : (s2 + r * DD);
        float acc = 0.0f;
        #pragma unroll
        for (int k = 0; k < DD; ++k) { float x = p[k]; acc = fmaf(x, x, acc); }
        sq[wave][lane >> 4][r] = acc;
    }
    __syncthreads();

    // --- WMMA accumulation: cross = A(16x32) * B^T(32x16), fp32
    // A 16x4 layout: lane L holds row m=L&15; VGPR0 = K(base+0), VGPR1 = K(base+1),
    // with base = 2*(L>=16). B 4x16 mirrors with col n=L&15.
    const int hi = lane >> 4;
    const int mn = lane & 15;
    const float* arow = s1 + mn * DD + hi * 2;
    const float* brow = s2 + mn * DD + hi * 2;

    v8f c = {};
    #pragma unroll
    for (int kk = 0; kk < 8; ++kk) {
        v2f a, bv;
        a.x  = arow[kk * 4 + 0];
        a.y  = arow[kk * 4 + 1];
        bv.x = brow[kk * 4 + 0];
        bv.y = brow[kk * 4 + 1];
        // (neg_a, A, neg_b, B, c_mod, C, reuse_a, reuse_b)
        c = __builtin_amdgcn_wmma_f32_16x16x4_f32(
                false, a, false, bv, (short)0, c, false, false);
    }

    // --- epilogue: d2 = sq1[m] + sq2[n] - 2*cross; cost = sqrt(max(d2,1e-12))
    // C/D layout: reg v -> row m = v + 8*hi, col n = lane&15.
    const float sq2n = sq[wave][1][mn];
    float* outp = cost + (size_t)b * NN * MM + (size_t)(tj * 16 + mn) * NN + (size_t)ti * 16;
    #pragma unroll
    for (int v = 0; v < 8; ++v) {
        const int m = v + hi * 8;
        float d2 = sq[wave][0][m] + sq2n - 2.0f * c[v];
        d2 = fmaxf(d2, 1e-12f);
        outp[m] = sqrtf(d2);
    }
}

// ---------------------------------------------------------------------------
// Kernel 2: DTW DP, one workgroup per batch. 512 threads = one per column j.
// Anti-diagonal wavefront: at step d, thread j computes cell (i=d-j, j).
// dp[i][j] = cost[i][j] + min(dp[i-1][j], dp[i][j-1], dp[i-1][j-1]),
// with dp[-1][-1]=0 and other out-of-range = +inf.  Three rotating diagonal
// buffers in LDS; slot(d) = d%3 holds diagonal d (overwrites d-3, which was
// last read at step d-1 -> single barrier per step after the write suffices).
// ---------------------------------------------------------------------------
__global__ __launch_bounds__(512) void dtw_dp_kernel(
    const float* __restrict__ cost,
    float* __restrict__ out)
{
    const int b = blockIdx.x;
    const int j = threadIdx.x; // column
    const float* col = cost + (size_t)b * NN * MM + (size_t)j * NN; // contiguous in i

    __shared__ float buf[3][MM];
    const float INF = __builtin_inff();

    buf[0][j] = INF;
    buf[1][j] = INF;
    buf[2][j] = INF;
    __syncthreads();

    #pragma unroll 1
    for (int d = 0; d < NN + MM - 1; ++d) {
        const int s0 = d % 3;            // slot for diagonal d (was d-3)
        const int s1 = (d + 2) % 3;      // slot holding diagonal d-1
        const int s2 = (d + 1) % 3;      // slot holding diagonal d-2

        const int i = d - j;
        float v = INF;
        if (i >= 0 && i < NN) {
            const float up   = (i >= 1) ? buf[s1][j] : INF;               // dp[i-1][j]
            const float left = (j >= 1) ? buf[s1][j - 1] : INF;           // dp[i][j-1]
            float diag;                                                   // dp[i-1][j-1]
            if (i >= 1 && j >= 1)      diag = buf[s2][j - 1];
            else if (i == 0 && j == 0) diag = 0.0f;                        // dp[-1][-1]
            else                       diag = INF;

            if (i + 32 < NN) __builtin_prefetch(col + i + 32, 0, 0);
            const float cst = col[i];
            v = cst + fminf(diag, fminf(up, left));
            if (i == NN - 1 && j == MM - 1) out[b] = v;   // dp[N-1][M-1]
        }
        buf[s0][j] = v;
        __syncthreads();
    }
}

extern "C" void kernel_launch(void* const* d_in, const int* in_sizes, int n_in,
                              void* d_out, int out_size, void* d_ws, size_t ws_size,
                              hipStream_t stream) {
    const float* seq1 = (const float*)d_in[0]; // (96, 512, 32) fp32
    const float* seq2 = (const float*)d_in[1]; // (96, 512, 32) fp32
    float* out  = (float*)d_out;               // (96,) fp32
    float* cost = (float*)d_ws;                // 96*512*512*4 = 96 MB scratch

    // Kernel 1: 96*32*32 = 98304 tiles, 8 waves/block -> 12288 blocks x 256
    dtw_cost_kernel<<<12288, 256, 0, stream>>>(seq1, seq2, cost);

    // Kernel 2: one block per batch, 512 threads (one per column)
    dtw_dp_kernel<<<BB, 512, 0, stream>>>(cost, out);
}